// MultiLayerGAT_14628658610377
// MI455X (gfx1250) — compile-verified
//
#include <hip/hip_runtime.h>
#include <math.h>

typedef float v2f __attribute__((ext_vector_type(2)));
typedef float v8f __attribute__((ext_vector_type(8)));

#define NN 20000
#define DD 128
#define HH 4
#define CC 32
#define NEG_SLOPE 0.2f
#define EPS_SM 1e-16f
#define LN_EPS 1e-5f

// ---------------------------------------------------------------------------
// float atomic max via sign-split integer punning (works for +/- values)
// ---------------------------------------------------------------------------
__device__ __forceinline__ void atomicMaxF32(float* addr, float v) {
    if (v >= 0.0f) atomicMax(reinterpret_cast<int*>(addr), __float_as_int(v));
    else           atomicMin(reinterpret_cast<unsigned int*>(addr), __float_as_uint(v));
}

// ---------------------------------------------------------------------------
// h = X @ W  via V_WMMA_F32_16X16X4_F32 (fp32-exact).
// Block = 256 threads = 8 waves; block -> one 16-row strip, wave -> one 16-col
// tile. A 16x4 frag: lanes 0-15 hold K={k0,k0+1}, lanes 16-31 K={k0+2,k0+3}.
// B 4x16 frag mirrors A with N across lanes. C/D: VGPR j -> row j + 8*half.
// ---------------------------------------------------------------------------
__global__ __launch_bounds__(256) void gat_gemm_wmma(
    const float* __restrict__ X, const float* __restrict__ W,
    float* __restrict__ Hout)
{
    const int lane = threadIdx.x & 31;
    const int wv   = threadIdx.x >> 5;        // 0..7 -> column tile
    const int row0 = blockIdx.x << 4;
    const int col0 = wv << 4;
    const int r    = lane & 15;
    const int hlf  = lane >> 4;

    v8f acc = {};
    const float* __restrict__ xrow = X + (size_t)(row0 + r) * DD;

    #pragma unroll
    for (int k0 = 0; k0 < DD; k0 += 4) {
        const int ka = k0 + 2 * hlf;
        v2f a, b;
        a.x = xrow[ka];
        a.y = xrow[ka + 1];
        b.x = W[(size_t)ka * DD + col0 + r];
        b.y = W[(size_t)(ka + 1) * DD + col0 + r];
        acc = __builtin_amdgcn_wmma_f32_16x16x4_f32(
            /*neg_a=*/false, a, /*neg_b=*/false, b,
            /*c_mod=*/(short)0, acc, /*reuse_a=*/false, /*reuse_b=*/false);
    }

    #pragma unroll
    for (int j = 0; j < 8; ++j)
        Hout[(size_t)(row0 + j + 8 * hlf) * DD + col0 + r] = acc[j];
}

// ---------------------------------------------------------------------------
// alpha_src / alpha_dst : one thread per (node, head)
// ---------------------------------------------------------------------------
__global__ void gat_alpha(const float* __restrict__ Hm,
                          const float* __restrict__ a_s,
                          const float* __restrict__ a_d,
                          float* __restrict__ AS, float* __restrict__ AD)
{
    int t = blockIdx.x * blockDim.x + threadIdx.x;   // n*H + h
    if (t >= NN * HH) return;
    int h = t & (HH - 1);
    int n = t >> 2;
    const float* hv = Hm + (size_t)n * DD + h * CC;
    const float* vs = a_s + h * CC;
    const float* vd = a_d + h * CC;
    float ss = 0.f, sd = 0.f;
    #pragma unroll
    for (int c = 0; c < CC; ++c) {
        float x = hv[c];
        ss += x * vs[c];
        sd += x * vd[c];
    }
    AS[t] = ss;
    AD[t] = sd;
}

// ---------------------------------------------------------------------------
// zero AGG, init EMAX=-inf, DEN=0
// ---------------------------------------------------------------------------
__global__ void gat_init(float* __restrict__ AGG, float* __restrict__ EMAX,
                         float* __restrict__ DEN)
{
    int i = blockIdx.x * blockDim.x + threadIdx.x;
    if (i < NN * DD) AGG[i] = 0.0f;
    if (i < NN * HH) {
        EMAX[i] = -__builtin_huge_valf();
        DEN[i]  = 0.0f;
    }
}

// ---------------------------------------------------------------------------
// pass 1: segment max of leaky-relu logits over destination
// ---------------------------------------------------------------------------
__global__ void gat_edge_max(const int* __restrict__ src,
                             const int* __restrict__ dst, int E,
                             const float* __restrict__ AS,
                             const float* __restrict__ AD,
                             float* __restrict__ EMAX)
{
    int e = blockIdx.x * blockDim.x + threadIdx.x;
    int EE = E + NN;
    if (e >= EE) return;
    int s, d;
    if (e < E) { s = src[e]; d = dst[e]; } else { s = e - E; d = s; }
    #pragma unroll
    for (int h = 0; h < HH; ++h) {
        float v = AS[s * HH + h] + AD[d * HH + h];
        v = (v > 0.f) ? v : NEG_SLOPE * v;
        atomicMaxF32(&EMAX[d * HH + h], v);
    }
}

// ---------------------------------------------------------------------------
// pass 2: segment sum of exp(e - emax[dst])
// ---------------------------------------------------------------------------
__global__ void gat_edge_sum(const int* __restrict__ src,
                             const int* __restrict__ dst, int E,
                             const float* __restrict__ AS,
                             const float* __restrict__ AD,
                             const float* __restrict__ EMAX,
                             float* __restrict__ DEN)
{
    int e = blockIdx.x * blockDim.x + threadIdx.x;
    int EE = E + NN;
    if (e >= EE) return;
    int s, d;
    if (e < E) { s = src[e]; d = dst[e]; } else { s = e - E; d = s; }
    #pragma unroll
    for (int h = 0; h < HH; ++h) {
        float v = AS[s * HH + h] + AD[d * HH + h];
        v = (v > 0.f) ? v : NEG_SLOPE * v;
        atomicAdd(&DEN[d * HH + h], expf(v - EMAX[d * HH + h]));
    }
}

// ---------------------------------------------------------------------------
// pass 3: weighted message scatter. One wave (32 lanes) per edge: lane = c,
// loop over H heads -> 128 fp32 atomics per edge, resolved in L2.
// ---------------------------------------------------------------------------
__global__ __launch_bounds__(256) void gat_edge_scatter(
    const int* __restrict__ src, const int* __restrict__ dst, int E,
    const float* __restrict__ AS, const float* __restrict__ AD,
    const float* __restrict__ EMAX, const float* __restrict__ DEN,
    const float* __restrict__ Hm, float* __restrict__ AGG)
{
    int lane = threadIdx.x & 31;
    int e = (blockIdx.x << 3) + (threadIdx.x >> 5);
    int EE = E + NN;
    if (e >= EE) return;
    int s, d;
    if (e < E) { s = src[e]; d = dst[e]; } else { s = e - E; d = s; }

    float w[HH];
    #pragma unroll
    for (int h = 0; h < HH; ++h) {
        float v = AS[s * HH + h] + AD[d * HH + h];
        v = (v > 0.f) ? v : NEG_SLOPE * v;
        float ex = expf(v - EMAX[d * HH + h]);
        w[h] = ex / (DEN[d * HH + h] + EPS_SM);
    }

    const float* __restrict__ hs = Hm + (size_t)s * DD;
    float* __restrict__ od = AGG + (size_t)d * DD;
    #pragma unroll
    for (int h = 0; h < HH; ++h) {
        int idx = h * CC + lane;
        atomicAdd(&od[idx], hs[idx] * w[h]);
    }
}

// ---------------------------------------------------------------------------
// bias + LayerNorm + ReLU, one wave32 per node, float4 lanes, shfl_xor reduce
// ---------------------------------------------------------------------------
__global__ __launch_bounds__(256) void gat_bias_ln_relu(
    const float* __restrict__ AGG, const float* __restrict__ bias,
    const float* __restrict__ g, const float* __restrict__ be,
    float* __restrict__ OUT)
{
    int lane = threadIdx.x & 31;
    int n = (blockIdx.x << 3) + (threadIdx.x >> 5);
    if (n >= NN) return;

    int base = lane * 4;
    float4 v = *reinterpret_cast<const float4*>(AGG + (size_t)n * DD + base);
    v.x += bias[base];     v.y += bias[base + 1];
    v.z += bias[base + 2]; v.w += bias[base + 3];

    float s = v.x + v.y + v.z + v.w;
    float q = v.x * v.x + v.y * v.y + v.z * v.z + v.w * v.w;
    #pragma unroll
    for (int m = 16; m >= 1; m >>= 1) {
        s += __shfl_xor(s, m, 32);
        q += __shfl_xor(q, m, 32);
    }
    float mean = s * (1.0f / DD);
    float var  = q * (1.0f / DD) - mean * mean;
    float inv  = rsqrtf(var + LN_EPS);

    float4 o;
    o.x = fmaxf((v.x - mean) * inv * g[base    ] + be[base    ], 0.f);
    o.y = fmaxf((v.y - mean) * inv * g[base + 1] + be[base + 1], 0.f);
    o.z = fmaxf((v.z - mean) * inv * g[base + 2] + be[base + 2], 0.f);
    o.w = fmaxf((v.w - mean) * inv * g[base + 3] + be[base + 3], 0.f);
    *reinterpret_cast<float4*>(OUT + (size_t)n * DD + base) = o;
}

// ---------------------------------------------------------------------------
extern "C" void kernel_launch(void* const* d_in, const int* in_sizes, int n_in,
                              void* d_out, int out_size, void* d_ws, size_t ws_size,
                              hipStream_t stream)
{
    (void)n_in; (void)out_size; (void)ws_size;

    const float* feat = (const float*)d_in[0];
    const int*   ei   = (const int*)d_in[1];
    const int    E    = in_sizes[1] / 2;
    const int*   src  = ei;
    const int*   dst  = ei + E;

    const float* W [2] = {(const float*)d_in[2],  (const float*)d_in[8]};
    const float* as_[2]= {(const float*)d_in[3],  (const float*)d_in[9]};
    const float* ad_[2]= {(const float*)d_in[4],  (const float*)d_in[10]};
    const float* bs[2] = {(const float*)d_in[5],  (const float*)d_in[11]};
    const float* gm[2] = {(const float*)d_in[6],  (const float*)d_in[12]};
    const float* bt[2] = {(const float*)d_in[7],  (const float*)d_in[13]};

    float* ws   = (float*)d_ws;
    float* X    = ws;                  // N*D  : layer input buffer
    float* Hm   = X    + (size_t)NN * DD;  // N*D  : per-head features
    float* AGG  = Hm   + (size_t)NN * DD;  // N*D  : aggregation
    float* AS   = AGG  + (size_t)NN * DD;  // N*H
    float* AD   = AS   + (size_t)NN * HH;  // N*H
    float* EMAX = AD   + (size_t)NN * HH;  // N*H
    float* DEN  = EMAX + (size_t)NN * HH;  // N*H

    const int EE = E + NN;
    dim3 blk(256);

    for (int b = 0; b < 2; ++b) {
        for (int l = 0; l < 2; ++l) {
            const float* xin  = (l == 0) ? feat + (size_t)b * NN * DD : X;
            float*       yout = (l == 0) ? X : (float*)d_out + (size_t)b * NN * DD;

            gat_gemm_wmma   <<<NN / 16,              256, 0, stream>>>(xin, W[l], Hm);
            gat_alpha       <<<(NN * HH + 255) / 256, blk, 0, stream>>>(Hm, as_[l], ad_[l], AS, AD);
            gat_init        <<<(NN * DD + 255) / 256, blk, 0, stream>>>(AGG, EMAX, DEN);
            gat_edge_max    <<<(EE + 255) / 256,      blk, 0, stream>>>(src, dst, E, AS, AD, EMAX);
            gat_edge_sum    <<<(EE + 255) / 256,      blk, 0, stream>>>(src, dst, E, AS, AD, EMAX, DEN);
            gat_edge_scatter<<<(EE + 7) / 8,          blk, 0, stream>>>(src, dst, E, AS, AD, EMAX, DEN, Hm, AGG);
            gat_bias_ln_relu<<<(NN + 7) / 8,          blk, 0, stream>>>(AGG, bs[l], gm[l], bt[l], yout);
        }
    }
}